// SCAGIA_17506286699088
// MI455X (gfx1250) — compile-verified
//
#include <hip/hip_runtime.h>
#include <math.h>
#include <stdint.h>

// ---------------------------------------------------------------------------
// Problem constants (B=4, C=512, H=W=128)
// ---------------------------------------------------------------------------
#define BN   4
#define CC   512
#define C2   256
#define C4   128
#define HH   128
#define WWID 128
#define HW   16384            // H*W
#define NQ   4096             // pooled spatial (64*64)
#define OUTSZ ((size_t)BN * CC * HW)   // per-image output floats

// LDS pitches: PA=20 for async-b128-staged A-tiles (16B-aligned float4 rows,
// conflict-free b64 frag reads); PB=18 for transposed B-tiles (conflict-free
// ds stores, 8B-aligned b64 frag reads).
#define PA 20
#define PB 18

typedef float v2f __attribute__((ext_vector_type(2)));
typedef float v4f __attribute__((ext_vector_type(4)));
typedef float v8f __attribute__((ext_vector_type(8)));

__device__ __forceinline__ float gelu_exact(float x) {
    // jax.nn.gelu(approximate=False): 0.5*x*(1+erf(x/sqrt(2)))
    return 0.5f * x * (1.0f + erff(x * 0.70710678118654752440f));
}

__device__ __forceinline__ v8f v8f_zero() {
    v8f z = {0.f, 0.f, 0.f, 0.f, 0.f, 0.f, 0.f, 0.f};
    return z;
}

// D = A(16x4, fp32) * B(4x16, fp32) + C(16x16, fp32)
__device__ __forceinline__ v8f wmma_f32(v2f a, v2f b, v8f c) {
    return __builtin_amdgcn_wmma_f32_16x16x4_f32(
        false, a, false, b, (short)0, c, false, false);
}

// Async global->LDS 16B-per-lane copy (ASYNCcnt-tracked CDNA5 path).
// LDS aperture flat address: low 32 bits == LDS byte offset.
__device__ __forceinline__ void async_load_b128(const void* g, void* l) {
    asm volatile("global_load_async_to_lds_b128 %0, %1, off"
                 :
                 : "v"((uint32_t)(uintptr_t)l), "v"((uint64_t)(uintptr_t)g)
                 : "memory");
}

__device__ __forceinline__ void wait_asynccnt0() {
    asm volatile("s_wait_asynccnt 0x0" ::: "memory");
}
__device__ __forceinline__ void wait_asynccnt2() {
    asm volatile("s_wait_asynccnt 0x2" ::: "memory");
}

// Workgroup barrier that drains only DScnt (LDS stores) — does NOT wait on
// LOADcnt, so in-flight global loads for the next pipeline stage stay live.
__device__ __forceinline__ void wg_barrier_dswait() {
    asm volatile("s_wait_dscnt 0x0\n"
                 "s_barrier_signal -1\n"
                 "s_barrier_wait -1" ::: "memory");
}
// Bare barrier: all LDS reads feeding this phase were already retired by the
// dscnt waits preceding the WMMAs that consumed them.
__device__ __forceinline__ void wg_barrier() {
    asm volatile("s_barrier_signal -1\n"
                 "s_barrier_wait -1" ::: "memory");
}

// ---------------------------------------------------------------------------
// Kernel 1: depthwise 3x3 conv + bias + exact GELU + cross-image channel swap
// Writes output channels [0, 256) of out1/out2.
// ---------------------------------------------------------------------------
__global__ __launch_bounds__(256) void dwconv_kernel(
    const float* __restrict__ x1, const float* __restrict__ x2,
    const float* __restrict__ dw_w, const float* __restrict__ dw_b,
    float* __restrict__ out)
{
    size_t idx = (size_t)blockIdx.x * 256 + threadIdx.x;
    int n    = (int)(idx & (HW - 1));
    size_t r = idx >> 14;
    int c    = (int)(r & (C2 - 1));  r >>= 8;
    int bt   = (int)(r & 3);
    int img  = (int)(r >> 2);

    const float* x = (img ? x2 : x1) + ((size_t)bt * CC + c) * HW;
    int h = n >> 7, w = n & 127;

    float s = 0.f;
    #pragma unroll
    for (int dh = -1; dh <= 1; dh++) {
        int hh = h + dh;
        if (hh < 0 || hh > 127) continue;
        #pragma unroll
        for (int dw = -1; dw <= 1; dw++) {
            int ww = w + dw;
            if (ww < 0 || ww > 127) continue;
            s += dw_w[c * 9 + (dh + 1) * 3 + (dw + 1)] * x[hh * WWID + ww];
        }
    }
    s = gelu_exact(s + dw_b[c]);

    int oimg = (c < C4) ? img : (1 - img);
    out[(size_t)oimg * OUTSZ + ((size_t)bt * CC + c) * HW + n] = s;
}

// ---------------------------------------------------------------------------
// Kernel 2: qkvl 1x1 conv as WMMA GEMM + bias + exact GELU.
// Grid: x = N/64 tiles, y = M/128, z = img*4+b. 8 waves/WG.
// Double-buffered LDS pipeline with manual barriers:
//   - W tile (i+1): async global->LDS b128 into sW[nxt] (s_wait_asynccnt 2)
//   - X tile (i+1): one global b128/thread into regs (stays in flight across
//     the barrier), transposed-stored next iteration.
// ---------------------------------------------------------------------------
__global__ __launch_bounds__(256) void qkvl_gemm_kernel(
    const float* __restrict__ x1, const float* __restrict__ x2,
    const float* __restrict__ wq, const float* __restrict__ bias,
    float* __restrict__ f1, float* __restrict__ f2)
{
    __shared__ float sW[2][128][PA];   // [buf][m][k]
    __shared__ float sXT[2][64][PB];   // [buf][n][k] (transposed)

    const int t    = threadIdx.x;
    const int wave = t >> 5;
    const int lane = t & 31;
    const int l16  = lane & 15;
    const int hi   = lane >> 4;

    const int n0  = blockIdx.x * 64;
    const int m0  = blockIdx.y * 128;
    const int ib  = blockIdx.z;
    const int img = ib >> 2;
    const int bt  = ib & 3;

    const float* x = (img ? x2 : x1) + ((size_t)bt * CC + C2) * HW;  // x[b, 256:, :]
    float*       f = (img ? f2 : f1) + (size_t)bt * CC * HW;

    // staging coordinates
    const int xrr = t >> 4;             // X tile row (k), 0..15
    const int xc4 = (t & 15) << 2;      // X tile col group (n), 0..60

    v8f acc[4];
    #pragma unroll
    for (int i = 0; i < 4; i++) acc[i] = v8f_zero();

    // ---- prologue: tile 0 ----
    #pragma unroll
    for (int j = 0; j < 2; j++) {
        int e = t + 256 * j;
        int rr = e >> 2, c4 = (e & 3) << 2;
        async_load_b128(&wq[(size_t)(m0 + rr) * C2 + c4], &sW[0][rr][c4]);
    }
    v4f xv = *(const v4f*)&x[(size_t)xrr * HW + n0 + xc4];

    const int NIT = C2 / 16;            // 16 k-steps
    for (int it = 0; it < NIT; ++it) {
        const int cur = it & 1, nxt = cur ^ 1;

        // store current X tile (transposed); waits for xv's global load only
        sXT[cur][xc4 + 0][xrr] = xv[0];
        sXT[cur][xc4 + 1][xrr] = xv[1];
        sXT[cur][xc4 + 2][xrr] = xv[2];
        sXT[cur][xc4 + 3][xrr] = xv[3];

        if (it + 1 < NIT) {
            const int k0n = (it + 1) * 16;
            // issue async W loads for next tile
            #pragma unroll
            for (int j = 0; j < 2; j++) {
                int e = t + 256 * j;
                int rr = e >> 2, c4 = (e & 3) << 2;
                async_load_b128(&wq[(size_t)(m0 + rr) * C2 + k0n + c4],
                                &sW[nxt][rr][c4]);
            }
            // issue global X load for next tile (consumed next iteration;
            // remains in flight across the barrier)
            xv = *(const v4f*)&x[(size_t)(k0n + xrr) * HW + n0 + xc4];
            if (k0n + 16 < C2)   // prefetch tile i+2 (global_prefetch_b8)
                __builtin_prefetch(&x[(size_t)(k0n + 16 + xrr) * HW + n0 + xc4], 0, 3);
            wait_asynccnt2();    // oldest 2 asyncs (current W tile) complete
        } else {
            wait_asynccnt0();
        }
        wg_barrier_dswait();     // drain DScnt only; xv load stays in flight

        #pragma unroll
        for (int kk = 0; kk < 16; kk += 4) {
            const int kp = kk + 2 * hi;
            v2f a = *(const v2f*)&sW[cur][wave * 16 + l16][kp];
            #pragma unroll
            for (int nt = 0; nt < 4; nt++) {
                v2f bb = *(const v2f*)&sXT[cur][nt * 16 + l16][kp];
                acc[nt] = wmma_f32(a, bb, acc[nt]);
            }
        }
        if (it + 1 < NIT) wg_barrier();   // protect next iter's buffer writes
    }

    const int mw = m0 + wave * 16;
    #pragma unroll
    for (int nt = 0; nt < 4; nt++) {
        #pragma unroll
        for (int r = 0; r < 8; r++) {
            int o = mw + r + hi * 8;
            int n = n0 + nt * 16 + l16;
            f[(size_t)o * HW + n] = gelu_exact(acc[nt][r] + bias[o]);
        }
    }
}

// ---------------------------------------------------------------------------
// Kernel 3: q = avgpool3x3 s2 p1 (/9, count_include_pad) of f[:,0];
//           k = maxpool2x2 s2 of f[:,1].
// ---------------------------------------------------------------------------
__global__ __launch_bounds__(256) void pool_kernel(
    const float* __restrict__ f1, const float* __restrict__ f2,
    float* __restrict__ qbuf, float* __restrict__ kbuf)
{
    size_t idx = (size_t)blockIdx.x * 256 + threadIdx.x;
    int p    = (int)(idx & (NQ - 1));
    size_t r = idx >> 12;
    int c    = (int)(r & (C4 - 1)); r >>= 7;
    int bt   = (int)(r & 3);        r >>= 2;
    int type = (int)(r & 1);
    int img  = (int)(r >> 1);

    int hq = p >> 6, wq = p & 63;
    const float* f = (img ? f2 : f1);

    if (type == 0) {
        const float* src = f + ((size_t)bt * CC + c) * HW;
        float s = 0.f;
        #pragma unroll
        for (int dh = -1; dh <= 1; dh++) {
            int hh = 2 * hq + dh;
            if (hh < 0 || hh > 127) continue;
            #pragma unroll
            for (int dw = -1; dw <= 1; dw++) {
                int ww = 2 * wq + dw;
                if (ww < 0 || ww > 127) continue;
                s += src[hh * WWID + ww];
            }
        }
        qbuf[(((size_t)img * BN + bt) * C4 + c) * NQ + p] = s * (1.0f / 9.0f);
    } else {
        const float* src = f + ((size_t)bt * CC + C4 + c) * HW;
        int hh = 2 * hq, ww = 2 * wq;
        float m = src[hh * WWID + ww];
        m = fmaxf(m, src[hh * WWID + ww + 1]);
        m = fmaxf(m, src[(hh + 1) * WWID + ww]);
        m = fmaxf(m, src[(hh + 1) * WWID + ww + 1]);
        kbuf[(((size_t)img * BN + bt) * C4 + c) * NQ + p] = m;
    }
}

// ---------------------------------------------------------------------------
// Kernel 4a: attention scores, split-K WMMA. Both tiles async-b128 staged.
// a[cq][ck] = sum_n q[cq][n]*k[ck][n] over an n-chunk of 512 per block.
// attends: 0:(q1,k1) 1:(q1,k2) 2:(q2,k2) 3:(q2,k1)
// ---------------------------------------------------------------------------
__global__ __launch_bounds__(256) void scores_kernel(
    const float* __restrict__ qbuf, const float* __restrict__ kbuf,
    float* __restrict__ partial)
{
    __shared__ float sQ[128][PA];   // [cq][k]
    __shared__ float sK[128][PA];   // [ck][k]

    const int t    = threadIdx.x;
    const int wave = t >> 5;
    const int lane = t & 31;
    const int l16  = lane & 15;
    const int hi   = lane >> 4;

    const int chunk = blockIdx.x;
    const int ab    = blockIdx.y;
    const int att   = ab >> 2;
    const int bt    = ab & 3;
    const int qimg  = att >> 1;
    const int kimg  = (att ^ (att >> 1)) & 1;

    const float* q = qbuf + ((size_t)qimg * BN + bt) * C4 * NQ;
    const float* k = kbuf + ((size_t)kimg * BN + bt) * C4 * NQ;

    v8f acc[8];
    #pragma unroll
    for (int i = 0; i < 8; i++) acc[i] = v8f_zero();

    const int kbeg = chunk * 512;
    for (int k0 = kbeg; k0 < kbeg + 512; k0 += 16) {
        #pragma unroll
        for (int j = 0; j < 2; j++) {
            int e = t + 256 * j;
            int rr = e >> 2, c4 = (e & 3) << 2;
            async_load_b128(&q[(size_t)rr * NQ + k0 + c4], &sQ[rr][c4]);
            async_load_b128(&k[(size_t)rr * NQ + k0 + c4], &sK[rr][c4]);
        }
        wait_asynccnt0();
        __syncthreads();
        #pragma unroll
        for (int kk = 0; kk < 16; kk += 4) {
            const int kp = kk + 2 * hi;
            v2f a = *(const v2f*)&sQ[wave * 16 + l16][kp];
            #pragma unroll
            for (int nt = 0; nt < 8; nt++) {
                v2f bb = *(const v2f*)&sK[nt * 16 + l16][kp];
                acc[nt] = wmma_f32(a, bb, acc[nt]);
            }
        }
        __syncthreads();
    }

    float* pp = partial + ((size_t)ab * 8 + chunk) * (C4 * C4);
    #pragma unroll
    for (int nt = 0; nt < 8; nt++) {
        #pragma unroll
        for (int r = 0; r < 8; r++) {
            int cq = wave * 16 + r + 8 * hi;
            int ck = nt * 16 + l16;
            pp[cq * C4 + ck] = acc[nt][r];
        }
    }
}

// ---------------------------------------------------------------------------
// Kernel 4b: reduce split-K partials; softmax over cq (axis=1) per column ck;
// store attn TRANSPOSED: attnT[ck][cq].
// ---------------------------------------------------------------------------
__global__ __launch_bounds__(256) void softmax_kernel(
    const float* __restrict__ partial, float* __restrict__ attnT)
{
    extern __shared__ float ssc[];          // [cq][ck] pitch 129
    const int t  = threadIdx.x;
    const int ab = blockIdx.x;
    const float* p = partial + (size_t)ab * 8 * (C4 * C4);

    for (int e = t; e < C4 * C4; e += 256) {
        float s = 0.f;
        #pragma unroll
        for (int ch = 0; ch < 8; ch++) s += p[(size_t)ch * (C4 * C4) + e];
        ssc[(e >> 7) * 129 + (e & 127)] = s;
    }
    __syncthreads();

    if (t < C4) {
        const int ck = t;
        float mx = -3.402823466e+38f;
        for (int cq = 0; cq < C4; cq++) mx = fmaxf(mx, ssc[cq * 129 + ck]);
        float sum = 0.f;
        for (int cq = 0; cq < C4; cq++) {
            float ev = expf(ssc[cq * 129 + ck] - mx);
            ssc[cq * 129 + ck] = ev;
            sum += ev;
        }
        const float inv = 1.0f / sum;
        float* a = attnT + (size_t)ab * (C4 * C4) + (size_t)ck * C4;
        for (int cq = 0; cq < C4; cq++) a[cq] = ssc[cq * 129 + ck] * inv;
    }
}

// ---------------------------------------------------------------------------
// Kernel 5: out[ck][n] = sum_cq attnT[ck][cq] * v[cq][n]. M=128,K=128,N=16384.
// A tile async-b128 staged; V tile: one global b128/thread -> transposed LDS.
// ---------------------------------------------------------------------------
__global__ __launch_bounds__(256) void apply_kernel(
    const float* __restrict__ attnT,
    const float* __restrict__ f1, const float* __restrict__ f2,
    float* __restrict__ out)
{
    __shared__ float sA[128][PA];   // [ck][cq]
    __shared__ float sVT[64][PB];   // [n][cq] (transposed)

    const int t    = threadIdx.x;
    const int wave = t >> 5;
    const int lane = t & 31;
    const int l16  = lane & 15;
    const int hi   = lane >> 4;

    const int n0  = blockIdx.x * 64;
    const int ab  = blockIdx.y;
    const int att = ab >> 2;
    const int bt  = ab & 3;
    const int img = att >> 1;
    const int vch = C2 + C4 * (att & 1);

    const float* v = (img ? f2 : f1) + ((size_t)bt * CC + vch) * HW;
    const float* A = attnT + (size_t)ab * (C4 * C4);
    float*       o = out + (size_t)img * OUTSZ + ((size_t)bt * CC + vch) * HW;

    const int xrr = t >> 4;             // V tile row (cq), 0..15
    const int xc4 = (t & 15) << 2;      // V tile col group (n), 0..60

    v8f acc[4];
    #pragma unroll
    for (int i = 0; i < 4; i++) acc[i] = v8f_zero();

    for (int k0 = 0; k0 < C4; k0 += 16) {
        #pragma unroll
        for (int j = 0; j < 2; j++) {
            int e = t + 256 * j;
            int rr = e >> 2, c4 = (e & 3) << 2;
            async_load_b128(&A[(size_t)rr * C4 + k0 + c4], &sA[rr][c4]);
        }
        v4f vv = *(const v4f*)&v[(size_t)(k0 + xrr) * HW + n0 + xc4];
        sVT[xc4 + 0][xrr] = vv[0];
        sVT[xc4 + 1][xrr] = vv[1];
        sVT[xc4 + 2][xrr] = vv[2];
        sVT[xc4 + 3][xrr] = vv[3];
        wait_asynccnt0();
        __syncthreads();
        #pragma unroll
        for (int kk = 0; kk < 16; kk += 4) {
            const int kp = kk + 2 * hi;
            v2f a = *(const v2f*)&sA[wave * 16 + l16][kp];
            #pragma unroll
            for (int nt = 0; nt < 4; nt++) {
                v2f bb = *(const v2f*)&sVT[nt * 16 + l16][kp];
                acc[nt] = wmma_f32(a, bb, acc[nt]);
            }
        }
        __syncthreads();
    }

    #pragma unroll
    for (int nt = 0; nt < 4; nt++) {
        #pragma unroll
        for (int r = 0; r < 8; r++) {
            int ck = wave * 16 + r + 8 * hi;
            int n  = n0 + nt * 16 + l16;
            o[(size_t)ck * HW + n] = acc[nt][r];
        }
    }
}

// ---------------------------------------------------------------------------
// Launch
// ---------------------------------------------------------------------------
extern "C" void kernel_launch(void* const* d_in, const int* in_sizes, int n_in,
                              void* d_out, int out_size, void* d_ws, size_t ws_size,
                              hipStream_t stream) {
    const float* x1     = (const float*)d_in[0];
    const float* x2     = (const float*)d_in[1];
    const float* dw_w   = (const float*)d_in[2];
    const float* dw_b   = (const float*)d_in[3];
    const float* qkvl_w = (const float*)d_in[4];
    const float* qkvl_b = (const float*)d_in[5];
    float* out = (float*)d_out;

    const size_t FSZ   = (size_t)BN * CC * HW;          // per-image f buffer
    const size_t QKSZ  = (size_t)2 * BN * C4 * NQ;
    float* f1      = (float*)d_ws;
    float* f2      = f1 + FSZ;
    float* qbuf    = f2 + FSZ;
    float* kbuf    = qbuf + QKSZ;
    float* partial = kbuf + QKSZ;                       // 16*8*16384 floats
    float* attnT   = partial + (size_t)16 * 8 * C4 * C4;

    dwconv_kernel<<<(2u * BN * C2 * HW) / 256, 256, 0, stream>>>(
        x1, x2, dw_w, dw_b, out);

    qkvl_gemm_kernel<<<dim3(HW / 64, CC / 128, 2 * BN), 256, 0, stream>>>(
        x1, x2, qkvl_w, qkvl_b, f1, f2);

    pool_kernel<<<(2u * 2 * BN * C4 * NQ) / 256, 256, 0, stream>>>(
        f1, f2, qbuf, kbuf);

    scores_kernel<<<dim3(8, 16), 256, 0, stream>>>(qbuf, kbuf, partial);
    softmax_kernel<<<16, 256, 128 * 129 * sizeof(float), stream>>>(partial, attnT);

    apply_kernel<<<dim3(HW / 64, 16), 256, 0, stream>>>(attnT, f1, f2, out);
}